// GraphTVLoss_80453327389002
// MI455X (gfx1250) — compile-verified
//
#include <hip/hip_runtime.h>
#include <hip/hip_bf16.h>
#include <math.h>

// GraphTV loss: mean_e( w_e * || x[src_e] - x[dst_e] ||_2 )
//
// Bandwidth-bound gather kernel. x (51.2 MB) fits in MI455X's 192 MB L2, so
// the goal is maximal concurrent global_load_b128 gathers. Row norms are
// computed as the diagonal of diff*diff^T via V_WMMA_F32_16X16X4_F32 (exact
// f32), using the A-fragment==B-fragment symmetry of Gram products.

typedef float v2f __attribute__((ext_vector_type(2)));
typedef float v8f __attribute__((ext_vector_type(8)));

#define WAVES_PER_BLOCK 4
#define THREADS (WAVES_PER_BLOCK * 32)
#define TILE_E 16            // edges per wave-tile (WMMA M dimension)
#define DVEC 128             // feature dim
#define DCOL 132             // padded LDS row stride: (4r+col)%64 banks -> conflict-free

__global__ __launch_bounds__(THREADS)
void tv_stage1(const float* __restrict__ x, const float* __restrict__ w,
               const int* __restrict__ src, const int* __restrict__ dst,
               float* __restrict__ wsum, int M)
{
    __shared__ float tile[WAVES_PER_BLOCK][TILE_E][DCOL];

    const int lane   = threadIdx.x & 31;
    const int wid    = threadIdx.x >> 5;
    const int gwave  = blockIdx.x * WAVES_PER_BLOCK + wid;
    const int nwaves = gridDim.x * WAVES_PER_BLOCK;
    const int ntiles = (M + TILE_E - 1) / TILE_E;

    float acc = 0.0f;

    for (int t = gwave; t < ntiles; t += nwaves) {
        const int base = t * TILE_E;

        // lanes 0..15 fetch edge metadata for edges base..base+15.
        // Out-of-range edges -> src=dst=0 (diff==0) and w=0 -> zero contribution.
        int   s16 = 0, d16 = 0;
        float w16 = 0.0f;
        if (lane < TILE_E) {
            const int e = base + lane;
            if (e < M) { s16 = src[e]; d16 = dst[e]; w16 = w[e]; }
        }

        // Gather both rows per edge (lane l covers dims 4l..4l+3 via b128),
        // difference in registers, stage diff tile in LDS.
        #pragma unroll 4
        for (int m = 0; m < TILE_E; ++m) {
            const int sm = __shfl(s16, m, 32);
            const int dm = __shfl(d16, m, 32);
            const float4 a = *(const float4*)(x + (size_t)sm * DVEC + 4 * lane);
            const float4 b = *(const float4*)(x + (size_t)dm * DVEC + 4 * lane);
            float4 df;
            df.x = a.x - b.x; df.y = a.y - b.y;
            df.z = a.z - b.z; df.w = a.w - b.w;
            *(float4*)&tile[wid][m][4 * lane] = df;
        }

        // Same-wave LDS ops are in-order (DScnt, ISA 7.3); just stop the
        // compiler from moving loads above the stores.
        asm volatile("" ::: "memory");

        // Gram diagonal: C += A * A^T over K=128 in 32 chunks of K=4.
        // A-fragment layout (16x4 f32): M = lane&15; lanes<16 hold K={0,1},
        // lanes>=16 hold K={2,3}. B-fragment of diff^T has identical per-lane
        // contents -> pass the same registers twice.
        v8f c = {0.f, 0.f, 0.f, 0.f, 0.f, 0.f, 0.f, 0.f};
        const int row   = lane & 15;
        const int khalf = (lane >> 4) << 1;   // 0 for lanes 0-15, 2 for 16-31
        #pragma unroll
        for (int kc = 0; kc < 32; ++kc) {
            const v2f frag = *(const v2f*)&tile[wid][row][4 * kc + khalf];
            c = __builtin_amdgcn_wmma_f32_16x16x4_f32(
                    false, frag, false, frag, (short)0, c, false, false);
        }

        // Diagonal extraction (C layout: VGPR r = rows r and r+8):
        //   lanes 0-7  hold diag m=lane    at c[lane]
        //   lanes 24-31 hold diag m=lane-16 at c[lane-24]
        const bool valid = (lane < 8) | (lane >= 24);
        const int  idx   = (lane < 8) ? lane : (lane - 24);
        float ss = 0.0f;
        #pragma unroll
        for (int r = 0; r < 8; ++r) {
            const float v = c[r];
            ss = (idx == r) ? v : ss;
        }
        // w for edge m lives in lane m (m = lane for lanes<8, lane-16 for >=24)
        const float wv = __shfl(w16, (lane < 8) ? lane : (lane - 16), 32);
        acc += valid ? wv * __builtin_sqrtf(ss) : 0.0f;
    }

    // Deterministic in-wave reduction, one partial per wave.
    #pragma unroll
    for (int off = 16; off > 0; off >>= 1)
        acc += __shfl_xor(acc, off, 32);
    if (lane == 0)
        wsum[gwave] = acc;
}

__global__ __launch_bounds__(256)
void tv_stage2(const float* __restrict__ ws, int n,
               float* __restrict__ out, float scale)
{
    __shared__ float sh[256];
    float s = 0.0f;
    for (int i = threadIdx.x; i < n; i += 256)   // fixed order -> deterministic
        s += ws[i];
    sh[threadIdx.x] = s;
    __syncthreads();
    for (int o = 128; o > 0; o >>= 1) {
        if ((int)threadIdx.x < o) sh[threadIdx.x] += sh[threadIdx.x + o];
        __syncthreads();
    }
    if (threadIdx.x == 0)
        out[0] = sh[0] * scale;   // ALPHA(=1) * mean
}

extern "C" void kernel_launch(void* const* d_in, const int* in_sizes, int n_in,
                              void* d_out, int out_size, void* d_ws, size_t ws_size,
                              hipStream_t stream)
{
    const float* x   = (const float*)d_in[0];
    const float* w   = (const float*)d_in[1];
    const int*   src = (const int*)d_in[2];
    const int*   dst = (const int*)d_in[3];
    const int    M   = in_sizes[1];

    int nblocks = 4096;                               // 16384 waves
    size_t need = (size_t)nblocks * WAVES_PER_BLOCK * sizeof(float);
    if (need > ws_size) {
        nblocks = (int)(ws_size / (WAVES_PER_BLOCK * sizeof(float)));
        if (nblocks < 1) nblocks = 1;
    }
    const int nwaves = nblocks * WAVES_PER_BLOCK;
    float* ws = (float*)d_ws;

    tv_stage1<<<nblocks, THREADS, 0, stream>>>(x, w, src, dst, ws, M);
    tv_stage2<<<1, 256, 0, stream>>>(ws, nwaves, (float*)d_out,
                                     1.0f / (float)M);
}